// MSDDAttn_88467736363077
// MI455X (gfx1250) — compile-verified
//
#include <hip/hip_runtime.h>
#include <hip/hip_bf16.h>

typedef float v2f __attribute__((ext_vector_type(2)));
typedef float v8f __attribute__((ext_vector_type(8)));

#define N_B   2
#define LQ    3060
#define LIN   3060
#define MROWS (N_B * LQ)     // 6120
#define DM    256

// -------------------------------------------------------------------------
// f32 GEMM via V_WMMA_F32_16X16X4_F32.
// C[M,Nc] = A[M,K] @ B[K,Nc] + bias[Nc].  K % 4 == 0, Nc % 32 == 0.
// One wave per 16x32 output tile (two independent WMMAs per k-step share
// one A fragment). M-edge handled with a clamped row + 0/1 mask so the
// inner loop has no EXEC manipulation and WMMA always runs with EXEC=~0.
// -------------------------------------------------------------------------
__global__ void gemm_wmma_f32(const float* __restrict__ A,
                              const float* __restrict__ B,
                              const float* __restrict__ bias,
                              float* __restrict__ C,
                              int M, int K, int Nc) {
    const int lane = threadIdx.x;          // 0..31
    const int wave = threadIdx.y;          // 0..7
    const int nTN2 = Nc >> 5;              // 16x32 tiles along N
    const int nTM  = (M + 15) >> 4;
    const long tile = (long)blockIdx.x * 8 + wave;      // wave-uniform
    if (tile >= (long)nTM * nTN2) return;               // whole wave exits
    const int tm = (int)(tile / nTN2);
    const int tn = (int)(tile % nTN2);

    const int mLane = lane & 15;
    const int kHalf = (lane >> 4) * 2;     // 0 or 2
    const int rowA  = tm * 16 + mLane;
    const int rowC  = (rowA < M) ? rowA : (M - 1);      // clamped, loop-invariant
    const float msk = (rowA < M) ? 1.0f : 0.0f;
    const int colB  = tn * 32 + mLane;

    const float* ap = A + (long)rowC * K + kHalf;       // +=4 per step
    const float* bp = B + (long)kHalf * Nc + colB;      // +=4*Nc per step

    v8f acc0 = {};
    v8f acc1 = {};
    #pragma unroll 2
    for (int k0 = 0; k0 < K; k0 += 4) {
        v2f a;
        a.x = ap[0] * msk;
        a.y = ap[1] * msk;
        v2f b0, b1;
        b0.x = bp[0];       b0.y = bp[Nc];
        b1.x = bp[16];      b1.y = bp[Nc + 16];
        acc0 = __builtin_amdgcn_wmma_f32_16x16x4_f32(
                   false, a, false, b0, (short)0, acc0, false, false);
        acc1 = __builtin_amdgcn_wmma_f32_16x16x4_f32(
                   false, a, false, b1, (short)0, acc1, false, false);
        ap += 4;
        bp += (long)4 * Nc;
    }

    const float bb0 = bias ? bias[colB]      : 0.0f;
    const float bb1 = bias ? bias[colB + 16] : 0.0f;
    const int rBase = tm * 16 + 8 * (lane >> 4);
    #pragma unroll
    for (int v = 0; v < 8; ++v) {
        int r = rBase + v;
        if (r < M) {
            float* cp = C + (long)r * Nc + colB;
            cp[0]  = acc0[v] + bb0;
            cp[16] = acc1[v] + bb1;
        }
    }
}

// -------------------------------------------------------------------------
// SE gate: x[n,ch] = mean_q value[n,q,ch]; probs[n,h,:] = softmax(x_h @ Wse + bse)
// grid = N_B blocks, 256 threads.
// -------------------------------------------------------------------------
__global__ void se_probs_kernel(const float* __restrict__ value,
                                const float* __restrict__ Wse,
                                const float* __restrict__ bse,
                                float* __restrict__ probs) {
    const int n = blockIdx.x;
    const int t = threadIdx.x;             // channel 0..255
    float s = 0.0f;
    const float* vp = value + (long)n * LIN * DM + t;
    for (int q = 0; q < LIN; ++q) s += vp[(long)q * DM];
    __shared__ float x[DM];
    x[t] = s * (1.0f / (float)LIN);
    __syncthreads();
    if (t < 8) {
        const int h = t;
        float z0 = bse[0], z1 = bse[1], z2 = bse[2];
        #pragma unroll
        for (int cc = 0; cc < 32; ++cc) {
            float xv = x[h * 32 + cc];
            z0 += xv * Wse[cc * 3 + 0];
            z1 += xv * Wse[cc * 3 + 1];
            z2 += xv * Wse[cc * 3 + 2];
        }
        float m  = fmaxf(z0, fmaxf(z1, z2));
        float e0 = expf(z0 - m), e1 = expf(z1 - m), e2 = expf(z2 - m);
        float inv = 1.0f / (e0 + e1 + e2);
        probs[(n * 8 + h) * 3 + 0] = e0 * inv;
        probs[(n * 8 + h) * 3 + 1] = e1 * inv;
        probs[(n * 8 + h) * 3 + 2] = e2 * inv;
    }
}

// -------------------------------------------------------------------------
// In-place softmax over 36 logits per (n,q,g,h). One thread per row.
// aw layout: [(n*LQ+q)*864 + g*288 + h*36 + j]
// -------------------------------------------------------------------------
__global__ void softmax36_kernel(float* __restrict__ aw) {
    const long rid = (long)blockIdx.x * blockDim.x + threadIdx.x;
    const long nrows = (long)MROWS * 24;   // (n,q) x (g,h)
    if (rid >= nrows) return;
    const long m  = rid / 24;              // n*LQ+q
    const int rem = (int)(rid % 24);
    const int g = rem / 8, h = rem % 8;
    float* p = aw + m * 864 + g * 288 + h * 36;
    float vals[36];
    float mx = -3.4e38f;
    #pragma unroll
    for (int j = 0; j < 36; ++j) { vals[j] = p[j]; mx = fmaxf(mx, vals[j]); }
    float s = 0.0f;
    #pragma unroll
    for (int j = 0; j < 36; ++j) { vals[j] = expf(vals[j] - mx); s += vals[j]; }
    const float inv = 1.0f / s;
    #pragma unroll
    for (int j = 0; j < 36; ++j) p[j] = vals[j] * inv;
}

// -------------------------------------------------------------------------
// Deformable aggregation. One wave per (n,q,g,h); lane = channel c (0..31).
// Only matmul positions j with k' = j//4 in {1,4,7} are nonzero:
//   j = (ki*3+1)*4 + l,  q' = q + (ki-1)*dilation,
//   sample = bilinear(value level l, head h, loc(n,q',g,h,l)),
//   weight = aw36[j]  (the reference's deliberate l/k flatten mismatch).
// Output: cat[(n*LQ+q)*768 + g*256 + h*32 + c], SE-gated.
// -------------------------------------------------------------------------
__global__ void aggregate_kernel(const float* __restrict__ value,
                                 const float* __restrict__ offb,
                                 const float* __restrict__ awb,
                                 const float* __restrict__ ref,
                                 const float* __restrict__ probs,
                                 float* __restrict__ cat) {
    const int c = threadIdx.x;             // 0..31
    const int h = threadIdx.y;             // 0..7
    const int b = blockIdx.x;              // N_B*LQ*3
    const int g = b % 3;
    const int t = b / 3;
    const int q = t % LQ;
    const int n = t / LQ;
    const int d = g + 1;                   // DILATIONS = (1,2,3)

    const int   WL[4] = {48, 24, 12, 6};
    const int   ST[4] = {0, 2304, 2880, 3024};
    const float FW[4] = {48.f, 24.f, 12.f, 6.f};

    const float* aw36 = awb + (long)(n * LQ + q) * 864 + g * 288 + h * 36;
    float acc = 0.0f;

    #pragma unroll
    for (int ki = 0; ki < 3; ++ki) {
        const int qp = q + (ki - 1) * d;
        if (qp < 0 || qp >= LQ) continue;
        const int jbase = (ki * 3 + 1) * 4;                  // 4, 16, 28
        const float* offq = offb + (long)(n * LQ + qp) * 192 + (g * 8 + h) * 8;
        const float* refq = ref  + (long)(n * LQ + qp) * 8;
        #pragma unroll
        for (int l = 0; l < 4; ++l) {
            const float w = aw36[jbase + l];
            const int   Wl = WL[l];
            const float Sf = FW[l];                          // W == H per level
            const float lx = refq[l * 2 + 0] + offq[l * 2 + 0] / Sf;
            const float ly = refq[l * 2 + 1] + offq[l * 2 + 1] / Sf;
            const float gx = lx * Sf - 0.5f;                 // align_corners=False
            const float gy = ly * Sf - 0.5f;
            const float x0f = floorf(gx), y0f = floorf(gy);
            const int x0 = (int)x0f, y0 = (int)y0f;
            const float wx1 = gx - x0f, wx0 = 1.0f - wx1;
            const float wy1 = gy - y0f, wy0 = 1.0f - wy1;
            const float* vb = value + ((long)n * LIN + ST[l]) * DM + h * 32 + c;
            float s = 0.0f;
            #pragma unroll
            for (int cy = 0; cy < 2; ++cy) {
                const int yi = y0 + cy;
                if (yi < 0 || yi >= Wl) continue;            // square levels
                const float wy = cy ? wy1 : wy0;
                #pragma unroll
                for (int cx = 0; cx < 2; ++cx) {
                    const int xi = x0 + cx;
                    if (xi < 0 || xi >= Wl) continue;
                    const float wx = cx ? wx1 : wx0;
                    s += wx * wy * vb[(long)(yi * Wl + xi) * DM];  // coalesced 128B/wave
                }
            }
            acc += w * s;
        }
    }
    acc *= probs[(n * 8 + h) * 3 + g];
    cat[(long)(n * LQ + q) * 768 + g * 256 + h * 32 + c] = acc;
}

// -------------------------------------------------------------------------
extern "C" void kernel_launch(void* const* d_in, const int* in_sizes, int n_in,
                              void* d_out, int out_size, void* d_ws, size_t ws_size,
                              hipStream_t stream) {
    const float* query = (const float*)d_in[0];   // (2,3060,256)
    const float* ref   = (const float*)d_in[1];   // (2,3060,4,2)
    const float* xin   = (const float*)d_in[2];   // (2,3060,256)
    const float* Wv    = (const float*)d_in[5];
    const float* bv    = (const float*)d_in[6];
    const float* Woff  = (const float*)d_in[7];
    const float* boff  = (const float*)d_in[8];
    const float* Wattn = (const float*)d_in[9];
    const float* battn = (const float*)d_in[10];
    const float* Wse   = (const float*)d_in[11];
    const float* bse   = (const float*)d_in[12];
    const float* Wout  = (const float*)d_in[13];
    const float* bout  = (const float*)d_in[14];
    float* out = (float*)d_out;

    float* ws    = (float*)d_ws;
    float* v_val = ws;                          // 6120*256  = 1,566,720
    float* v_off = v_val + (long)MROWS * 256;   // 6120*192  = 1,175,040
    float* v_aw  = v_off + (long)MROWS * 192;   // 6120*864  = 5,287,680
    float* v_cat = v_aw  + (long)MROWS * 864;   // 6120*768  = 4,700,160
    float* v_pr  = v_cat + (long)MROWS * 768;   // 48

    const dim3 gblk(32, 8);
    auto nblocks = [](long tiles) { return (int)((tiles + 7) / 8); };
    const int tM = (MROWS + 15) / 16;           // 383

    // value = X @ Wv + bv                       (tiles = 383 * 256/32)
    gemm_wmma_f32<<<nblocks((long)tM * 8), gblk, 0, stream>>>(
        xin, Wv, bv, v_val, MROWS, 256, 256);
    // SE gate probabilities
    se_probs_kernel<<<N_B, 256, 0, stream>>>(v_val, Wse, bse, v_pr);
    // off = Q @ Woff + boff                     (tiles = 383 * 192/32)
    gemm_wmma_f32<<<nblocks((long)tM * 6), gblk, 0, stream>>>(
        query, Woff, boff, v_off, MROWS, 256, 192);
    // aw = Q @ Wattn + battn                    (tiles = 383 * 864/32)
    gemm_wmma_f32<<<nblocks((long)tM * 27), gblk, 0, stream>>>(
        query, Wattn, battn, v_aw, MROWS, 256, 864);
    // softmax over 36 per (n,q,g,h)
    {
        const long rows = (long)MROWS * 24;
        softmax36_kernel<<<(int)((rows + 255) / 256), 256, 0, stream>>>(v_aw);
    }
    // deformable sampling + SE gating -> concat buffer
    aggregate_kernel<<<N_B * LQ * 3, gblk, 0, stream>>>(
        v_val, v_off, v_aw, ref, v_pr, v_cat);
    // out = cat @ Wout + bout                   (tiles = 383 * 256/32)
    gemm_wmma_f32<<<nblocks((long)tM * 8), gblk, 0, stream>>>(
        v_cat, Wout, bout, out, MROWS, 768, 256);
}